// DKVMN_38792144617767
// MI455X (gfx1250) — compile-verified
//
#include <hip/hip_runtime.h>
#include <hip/hip_bf16.h>
#include <math.h>

typedef __attribute__((ext_vector_type(2))) float v2f;
typedef __attribute__((ext_vector_type(8))) float v8f;

namespace {
constexpr int kNumQ = 1000;
constexpr int kD    = 64;     // DIM_S
constexpr int kM    = 50;     // SIZE_M
constexpr int kB    = 256;
constexpr int kN    = 200;
constexpr int kRows = kB * kN;            // 51200 (divisible by 64)

// workspace layout (floats)
constexpr size_t kOffW = 0;                              // [rows][50]
constexpr size_t kOffE = (size_t)kRows * kM;             // [rows][64]
constexpr size_t kOffA = kOffE + (size_t)kRows * kD;     // [rows][64]
constexpr size_t kOffR = kOffA + (size_t)kRows * kD;     // read [rows][64]
}

__device__ __forceinline__ v8f wmma_f32_k4(v2f a, v2f b, v8f c) {
  // D = A(16x4,f32) * B(4x16,f32) + C(16x16,f32)  -> v_wmma_f32_16x16x4_f32
  return __builtin_amdgcn_wmma_f32_16x16x4_f32(
      /*neg_a=*/false, a, /*neg_b=*/false, b,
      /*c_mod=*/(short)0, c, /*reuse_a=*/false, /*reuse_b=*/false);
}

// B-fragment for C[i,n] = sum_k A[i,k] * W[n,k]:  B[k][n] = W[n][k].
// With W staged row-major in LDS ([n][k], ldk = 64), the fragment
// {B[kb][n], B[kb+1][n]} = {W[n][kb], W[n][kb+1]} is one contiguous v2f.

// ---------------------------------------------------------------------------
// Phase 1: per 16-row tile -> w = softmax(k Mk^T), e = sigmoid(v eW^T),
//          a = tanh(v aW^T). One wave per tile, f32 WMMA, K=64.
// ---------------------------------------------------------------------------
__global__ __launch_bounds__(128)
void dkvmn_phase1(const int* __restrict__ q, const int* __restrict__ r,
                  const float* __restrict__ k_emb, const float* __restrict__ v_emb,
                  const float* __restrict__ Mk,
                  const float* __restrict__ eW, const float* __restrict__ eb,
                  const float* __restrict__ aW, const float* __restrict__ ab,
                  float* __restrict__ ws)
{
  __shared__ float sW[kD * kD];       // weight, row-major [n][k] (16 KB)
  __shared__ float sAk[4][16 * kD];   // per-wave k tile (reused for logits)
  __shared__ float sAv[4][16 * kD];   // per-wave v tile

  const int tid  = threadIdx.x;
  const int wave = tid >> 5;
  const int lane = tid & 31;
  const int hi   = lane >> 4;         // 0: K+0, 1: K+2 half of fragment
  const int idx  = lane & 15;
  const int rowBase = blockIdx.x * 64 + wave * 16;

  float* Ak = sAk[wave];
  float* Av = sAv[wave];

  // gather k / v rows of this wave's tile
  for (int i = 0; i < 16; ++i) {
    const int grow = rowBase + i;
    const int qi = q[grow];
    const int xi = qi + kNumQ * r[grow];
    for (int d0 = lane; d0 < kD; d0 += 32) {
      Ak[i * kD + d0] = k_emb[qi * kD + d0];
      Av[i * kD + d0] = v_emb[xi * kD + d0];
    }
  }

  // ---- GEMM 1: logits = k @ Mk^T (slots padded 50->64 with zeros) ----
  for (int i = tid; i < kD * kD; i += 128) {         // direct coalesced copy
    sW[i] = (i < kM * kD) ? Mk[i] : 0.0f;
  }
  __syncthreads();

  v8f acc[4];
  #pragma unroll
  for (int nt = 0; nt < 4; ++nt) acc[nt] = v8f{0,0,0,0,0,0,0,0};

  for (int kk = 0; kk < kD; kk += 4) {
    const int kb = kk + 2 * hi;
    const v2f af = *(const v2f*)&Ak[idx * kD + kb];
    #pragma unroll
    for (int nt = 0; nt < 4; ++nt) {
      const v2f bf = *(const v2f*)&sW[(nt * 16 + idx) * kD + kb];
      acc[nt] = wmma_f32_k4(af, bf, acc[nt]);
    }
  }
  // dump logits into Ak (k tile no longer needed)
  #pragma unroll
  for (int nt = 0; nt < 4; ++nt)
    #pragma unroll
    for (int rr = 0; rr < 8; ++rr)
      Ak[(rr + 8 * hi) * kD + nt * 16 + idx] = acc[nt][rr];
  __syncthreads();

  // softmax over first 50 columns, one lane per row
  if (lane < 16) {
    const int grow = rowBase + lane;
    float mx = -3.0e38f;
    for (int m = 0; m < kM; ++m) mx = fmaxf(mx, Ak[lane * kD + m]);
    float s = 0.0f;
    for (int m = 0; m < kM; ++m) {
      const float ev = expf(Ak[lane * kD + m] - mx);
      Ak[lane * kD + m] = ev;
      s += ev;
    }
    const float inv = 1.0f / s;
    for (int m = 0; m < kM; ++m)
      ws[kOffW + (size_t)grow * kM + m] = Ak[lane * kD + m] * inv;
  }
  __syncthreads();

  // ---- GEMM 2: e = sigmoid(v @ eW^T + eb) ----
  for (int i = tid; i < kD * kD; i += 128) sW[i] = eW[i];
  __syncthreads();

  #pragma unroll
  for (int nt = 0; nt < 4; ++nt) acc[nt] = v8f{0,0,0,0,0,0,0,0};
  for (int kk = 0; kk < kD; kk += 4) {
    const int kb = kk + 2 * hi;
    const v2f af = *(const v2f*)&Av[idx * kD + kb];
    #pragma unroll
    for (int nt = 0; nt < 4; ++nt) {
      const v2f bf = *(const v2f*)&sW[(nt * 16 + idx) * kD + kb];
      acc[nt] = wmma_f32_k4(af, bf, acc[nt]);
    }
  }
  #pragma unroll
  for (int nt = 0; nt < 4; ++nt) {
    const int col = nt * 16 + idx;
    const float bias = eb[col];
    #pragma unroll
    for (int rr = 0; rr < 8; ++rr) {
      const int grow = rowBase + rr + 8 * hi;
      const float x = acc[nt][rr] + bias;
      ws[kOffE + (size_t)grow * kD + col] = 1.0f / (1.0f + expf(-x));
    }
  }
  __syncthreads();

  // ---- GEMM 3: a = tanh(v @ aW^T + ab) ----
  for (int i = tid; i < kD * kD; i += 128) sW[i] = aW[i];
  __syncthreads();

  #pragma unroll
  for (int nt = 0; nt < 4; ++nt) acc[nt] = v8f{0,0,0,0,0,0,0,0};
  for (int kk = 0; kk < kD; kk += 4) {
    const int kb = kk + 2 * hi;
    const v2f af = *(const v2f*)&Av[idx * kD + kb];
    #pragma unroll
    for (int nt = 0; nt < 4; ++nt) {
      const v2f bf = *(const v2f*)&sW[(nt * 16 + idx) * kD + kb];
      acc[nt] = wmma_f32_k4(af, bf, acc[nt]);
    }
  }
  #pragma unroll
  for (int nt = 0; nt < 4; ++nt) {
    const int col = nt * 16 + idx;
    const float bias = ab[col];
    #pragma unroll
    for (int rr = 0; rr < 8; ++rr) {
      const int grow = rowBase + rr + 8 * hi;
      ws[kOffA + (size_t)grow * kD + col] = tanhf(acc[nt][rr] + bias);
    }
  }
}

// ---------------------------------------------------------------------------
// Phase 2: the serial memory scan. One workgroup per batch element; Mv state
// in registers (thread (mb,d) owns Mv[mb+4j, d]); read-before-write; streams
// Mv_{t+1} (12.8 KB/step) straight to d_out. This is the HBM-store roofline.
// ---------------------------------------------------------------------------
__global__ __launch_bounds__(256)
void dkvmn_phase2(const float* __restrict__ Mv0,
                  const float* __restrict__ ws_c,
                  float* __restrict__ wsr,
                  float* __restrict__ out)
{
  __shared__ float sw[kM];
  __shared__ float se[kD];
  __shared__ float sa[kD];
  __shared__ float part[256];

  const int b   = blockIdx.x;
  const int tid = threadIdx.x;
  const int d   = tid & 63;
  const int mb  = tid >> 6;    // 0..3

  float mv[13];
  #pragma unroll
  for (int j = 0; j < 13; ++j) {
    const int m = mb + 4 * j;
    mv[j] = (m < kM) ? Mv0[m * kD + d] : 0.0f;
  }

  float* MvOut = out + kRows;                       // Mv after p
  float* base0 = MvOut + (size_t)b * (kN + 1) * kM * kD;
  #pragma unroll
  for (int j = 0; j < 13; ++j) {                    // Mv[:,0,:,:] = Mv0
    const int m = mb + 4 * j;
    if (m < kM) base0[m * kD + d] = mv[j];
  }

  const float* wsw = ws_c + kOffW;
  const float* wse = ws_c + kOffE;
  const float* wsa = ws_c + kOffA;

  for (int t = 0; t < kN; ++t) {
    const size_t rrow = (size_t)b * kN + t;
    if (tid < kM)                      sw[tid]       = wsw[rrow * kM + tid];
    else if (tid >= 64 && tid < 128)   se[tid - 64]  = wse[rrow * kD + (tid - 64)];
    else if (tid >= 128 && tid < 192)  sa[tid - 128] = wsa[rrow * kD + (tid - 128)];
    if (tid == 0 && t + 1 < kN) {
      __builtin_prefetch(&wsw[(rrow + 1) * kM], 0, 0);   // global_prefetch_b8
      __builtin_prefetch(&wse[(rrow + 1) * kD], 0, 0);
      __builtin_prefetch(&wsa[(rrow + 1) * kD], 0, 0);
    }
    __syncthreads();

    const float ed = se[d];
    const float ad = sa[d];
    float pr = 0.0f;
    float* baseT = base0 + (size_t)(t + 1) * kM * kD;
    #pragma unroll
    for (int j = 0; j < 13; ++j) {
      const int m = mb + 4 * j;
      if (m < kM) {
        const float wm  = sw[m];
        const float old = mv[j];
        pr += wm * old;                               // read BEFORE write
        const float nv = old * (1.0f - wm * ed) + wm * ad;
        mv[j] = nv;
        baseT[m * kD + d] = nv;                       // stream Mv_{t+1}
      }
    }
    part[tid] = pr;
    __syncthreads();
    if (tid < kD) {
      const float rd = part[tid] + part[tid + 64] + part[tid + 128] + part[tid + 192];
      wsr[rrow * kD + tid] = rd;                      // read_t
    }
  }
}

// ---------------------------------------------------------------------------
// Phase 3: f = tanh([read|k] @ fW^T + fb); p = sigmoid(f . pW + pb).
// One wave per 16-row tile; K=128 done as two 64-wide passes (16 KB B slice).
// ---------------------------------------------------------------------------
__global__ __launch_bounds__(128)
void dkvmn_phase3(const int* __restrict__ q,
                  const float* __restrict__ k_emb,
                  const float* __restrict__ fW, const float* __restrict__ fb,
                  const float* __restrict__ pW, const float* __restrict__ pb,
                  const float* __restrict__ ws, float* __restrict__ out)
{
  __shared__ float sBW[kD * kD];         // fW slice, row-major [j][c] (16 KB)
  __shared__ float sA[4][16 * 2 * kD];   // per-wave [16][128]         (32 KB)
  __shared__ float sPW[kD];

  const int tid  = threadIdx.x;
  const int wave = tid >> 5;
  const int lane = tid & 31;
  const int hi   = lane >> 4;
  const int idx  = lane & 15;
  const int rowBase = blockIdx.x * 64 + wave * 16;
  float* A = sA[wave];
  if (tid < kD) sPW[tid] = pW[tid];

  const float* wsr = ws + kOffR;
  for (int i = 0; i < 16; ++i) {
    const int grow = rowBase + i;
    const int qi = q[grow];
    for (int d0 = lane; d0 < kD; d0 += 32) {
      A[i * 128 + d0]      = wsr[(size_t)grow * kD + d0];  // read
      A[i * 128 + kD + d0] = k_emb[qi * kD + d0];          // k
    }
  }

  v8f acc[4];
  #pragma unroll
  for (int nt = 0; nt < 4; ++nt) acc[nt] = v8f{0,0,0,0,0,0,0,0};

  for (int h = 0; h < 2; ++h) {
    __syncthreads();
    for (int i = tid; i < kD * kD; i += 128) {
      const int jj = i >> 6, cc = i & 63;
      sBW[i] = fW[jj * 128 + h * kD + cc];   // fW[j][h*64+c], row-major slice
    }
    __syncthreads();
    for (int kk = 0; kk < kD; kk += 4) {
      const int kb = kk + 2 * hi;
      const v2f af = *(const v2f*)&A[idx * 128 + h * kD + kb];
      #pragma unroll
      for (int nt = 0; nt < 4; ++nt) {
        const v2f bf = *(const v2f*)&sBW[(nt * 16 + idx) * kD + kb];
        acc[nt] = wmma_f32_k4(af, bf, acc[nt]);
      }
    }
  }
  __syncthreads();

  float* F = A;                          // reuse: f tile [16][64]
  #pragma unroll
  for (int nt = 0; nt < 4; ++nt) {
    const int col = nt * 16 + idx;
    const float bias = fb[col];
    #pragma unroll
    for (int rr = 0; rr < 8; ++rr)
      F[(rr + 8 * hi) * kD + col] = tanhf(acc[nt][rr] + bias);
  }
  __syncthreads();

  if (lane < 16) {
    const int grow = rowBase + lane;
    float s = pb[0];
    for (int j = 0; j < kD; ++j) s += F[lane * kD + j] * sPW[j];
    out[grow] = 1.0f / (1.0f + expf(-s));
  }
}

// ---------------------------------------------------------------------------
extern "C" void kernel_launch(void* const* d_in, const int* in_sizes, int n_in,
                              void* d_out, int out_size, void* d_ws, size_t ws_size,
                              hipStream_t stream) {
  (void)in_sizes; (void)n_in; (void)out_size; (void)ws_size;
  const int*   q     = (const int*)  d_in[0];
  const int*   r     = (const int*)  d_in[1];
  const float* k_emb = (const float*)d_in[2];
  const float* v_emb = (const float*)d_in[3];
  const float* Mk    = (const float*)d_in[4];
  const float* Mv0   = (const float*)d_in[5];
  const float* eW    = (const float*)d_in[6];
  const float* eb    = (const float*)d_in[7];
  const float* aW    = (const float*)d_in[8];
  const float* ab    = (const float*)d_in[9];
  const float* fW    = (const float*)d_in[10];
  const float* fb    = (const float*)d_in[11];
  const float* pW    = (const float*)d_in[12];
  const float* pb    = (const float*)d_in[13];

  float* out = (float*)d_out;
  float* ws  = (float*)d_ws;

  dkvmn_phase1<<<kRows / 64, 128, 0, stream>>>(q, r, k_emb, v_emb, Mk,
                                               eW, eb, aW, ab, ws);
  dkvmn_phase2<<<kB, 256, 0, stream>>>(Mv0, ws, ws + kOffR, out);
  dkvmn_phase3<<<kRows / 64, 128, 0, stream>>>(q, k_emb, fW, fb, pW, pb, ws, out);
}